// MultiheadSelfAttention_50818053046707
// MI455X (gfx1250) — compile-verified
//
#include <hip/hip_runtime.h>
#include <hip/hip_bf16.h>

typedef float v2f __attribute__((ext_vector_type(2)));
typedef float v8f __attribute__((ext_vector_type(8)));

#define BB    2
#define SS    4096
#define CC    2048
#define EPSF  1.1920929e-07f

#define BM    128    // M rows per workgroup tile
#define BN    128    // N cols per workgroup tile (== head dim)
#define BK    32     // K chunk staged in LDS
#define LDA   34     // padded LDS stride (8B-aligned, conflict-free)
#define LDE   130    // padded LDS stride for epilogue tile
#define BUFSZ (2 * BM * LDA)   // floats per double-buffer slot (A+B) = 8704
#define NCHUNK (CC / BK)       // 64

__global__ __launch_bounds__(256)
void qkv_rmsnorm_rope_kernel(const float* __restrict__ x,
                             const float* __restrict__ rope,
                             const float* __restrict__ W,
                             const float* __restrict__ bias,
                             const float* __restrict__ nw,
                             float* __restrict__ out,
                             int do_norm_rope)
{
    __shared__ float smem[2 * BUFSZ];     // 69,632 B: two (A|B) stages
    float* Ep  = smem;                    // [64][LDE] epilogue tile, aliases buf0
    float* Inv = smem + 64 * LDE;

    const int t     = threadIdx.x;
    const int wave  = t >> 5;
    const int lane  = t & 31;
    const int lr    = lane & 15;
    const int hoff  = (lane < 16) ? 0 : 8;   // C/D row offset, upper half-wave
    const int koff  = (lane < 16) ? 0 : 2;   // A/B K offset, upper half-wave
    const int waveM = wave & 3;              // 4 row-groups of 32
    const int waveN = wave >> 2;             // 2 col-groups of 64

    const int mbase = blockIdx.x * BM;
    const int nbase = blockIdx.y * BN;

    // ---- init accumulators with bias ----
    v8f acc[2][4];
#pragma unroll
    for (int j = 0; j < 4; ++j) {
        float bv = bias[nbase + waveN * 64 + j * 16 + lr];
#pragma unroll
        for (int i = 0; i < 2; ++i)
#pragma unroll
            for (int v = 0; v < 8; ++v)
                acc[i][j][v] = bv;
    }

    // fragment element offsets within one buffer
    int aoff[2], boff[4];
#pragma unroll
    for (int i = 0; i < 2; ++i) aoff[i] = (waveM * 32 + i * 16 + lr) * LDA + koff;
#pragma unroll
    for (int j = 0; j < 4; ++j) boff[j] = (BM * LDA) + (waveN * 64 + j * 16 + lr) * LDA + koff;

    // per-thread staging coords: 4 float4 rows-of-32 per matrix
    // async copy: global -> LDS, 16B per lane, tracked by ASYNCcnt
    auto issue_chunk = [&](int k0, int buf) {
        float* Ab = smem + buf * BUFSZ;
        float* Bb = Ab + BM * LDA;
#pragma unroll
        for (int it = 0; it < 4; ++it) {
            int c  = t + it * 256;
            int r  = c >> 3;
            int cg = (c & 7) << 2;
            unsigned la = (unsigned)(size_t)(&Ab[r * LDA + cg]);
            const float* ga = x + (size_t)(mbase + r) * CC + k0 + cg;
            asm volatile("global_load_async_to_lds_b128 %0, %1, off"
                         :: "v"(la), "v"(ga) : "memory");
            unsigned lb = (unsigned)(size_t)(&Bb[r * LDA + cg]);
            const float* gb = W + (size_t)(nbase + r) * CC + k0 + cg;
            asm volatile("global_load_async_to_lds_b128 %0, %1, off"
                         :: "v"(lb), "v"(gb) : "memory");
        }
    };

    // ---- software-pipelined main loop ----
    issue_chunk(0, 0);
    for (int kc = 0; kc < NCHUNK; ++kc) {
        const int cur = kc & 1;
        asm volatile("s_wait_asynccnt 0x0" ::: "memory");  // my stage-writes landed
        __syncthreads();                                   // everyone's landed
        if (kc + 1 < NCHUNK)
            issue_chunk((kc + 1) * BK, cur ^ 1);           // prefetch next stage

        const float* buf = smem + cur * BUFSZ;
#pragma unroll
        for (int kk = 0; kk < BK; kk += 4) {
            v2f a[2], b[4];
#pragma unroll
            for (int i = 0; i < 2; ++i)
                a[i] = *reinterpret_cast<const v2f*>(&buf[aoff[i] + kk]);
#pragma unroll
            for (int j = 0; j < 4; ++j)
                b[j] = *reinterpret_cast<const v2f*>(&buf[boff[j] + kk]);
#pragma unroll
            for (int i = 0; i < 2; ++i)
#pragma unroll
                for (int j = 0; j < 4; ++j)
                    acc[i][j] = __builtin_amdgcn_wmma_f32_16x16x4_f32(
                        false, a[i], false, b[j], (short)0, acc[i][j],
                        false, false);
        }
    }

    // ---- fused epilogue: two 64-row passes through LDS ----
    for (int half = 0; half < 2; ++half) {
        __syncthreads();
        if ((waveM >> 1) == half) {
            int rb = (waveM & 1) * 32;
#pragma unroll
            for (int i = 0; i < 2; ++i)
#pragma unroll
                for (int j = 0; j < 4; ++j) {
                    int col = waveN * 64 + j * 16 + lr;
                    int er  = rb + i * 16 + hoff;
#pragma unroll
                    for (int v = 0; v < 8; ++v)
                        Ep[(er + v) * LDE + col] = acc[i][j][v];
                }
        }
        __syncthreads();

        if (do_norm_rope) {
            if (t < 64) {
                float s = 0.0f;
                for (int c = 0; c < BN; ++c) {
                    float q = Ep[t * LDE + c];
                    s += q * q;
                }
                Inv[t] = rsqrtf(s * (1.0f / (float)BN) + EPSF);
            }
            __syncthreads();
        }

        // 64 rows x 64 pairs = 4096 items, 16 iters x 256 threads; coalesced writes
        for (int it = 0; it < 16; ++it) {
            int idx = it * 256 + t;
            int er  = idx >> 6;
            int p   = idx & 63;
            int m   = mbase + half * 64 + er;
            float t0 = Ep[er * LDE + 2 * p];
            float t1 = Ep[er * LDE + 2 * p + 1];
            float2 o;
            if (do_norm_rope) {
                int   sseq = m & (SS - 1);
                float inv  = Inv[er];
                t0 *= inv * nw[2 * p];
                t1 *= inv * nw[2 * p + 1];
                float4 r4 = *reinterpret_cast<const float4*>(
                    rope + (size_t)sseq * 256 + p * 4);   // [cos, -sin, sin, cos]
                o.x = r4.x * t0 + r4.y * t1;
                o.y = r4.z * t0 + r4.w * t1;
            } else {
                o.x = t0;
                o.y = t1;
            }
            *reinterpret_cast<float2*>(out + (size_t)m * CC + nbase + 2 * p) = o;
        }
        __syncthreads();
    }
}

extern "C" void kernel_launch(void* const* d_in, const int* in_sizes, int n_in,
                              void* d_out, int out_size, void* d_ws, size_t ws_size,
                              hipStream_t stream) {
    (void)in_sizes; (void)n_in; (void)out_size; (void)d_ws; (void)ws_size;
    const float* x    = (const float*)d_in[0];
    const float* rope = (const float*)d_in[1];
    const float* wq   = (const float*)d_in[2];
    const float* bq   = (const float*)d_in[3];
    const float* wk   = (const float*)d_in[4];
    const float* bk   = (const float*)d_in[5];
    const float* wv   = (const float*)d_in[6];
    const float* bv   = (const float*)d_in[7];
    const float* qn   = (const float*)d_in[8];
    const float* kn   = (const float*)d_in[9];
    float* out = (float*)d_out;

    const size_t BSC = (size_t)BB * SS * CC;
    dim3 grid(BB * SS / BM, CC / BN, 1);
    dim3 block(256, 1, 1);

    qkv_rmsnorm_rope_kernel<<<grid, block, 0, stream>>>(x, rope, wq, bq, qn, out, 1);
    qkv_rmsnorm_rope_kernel<<<grid, block, 0, stream>>>(x, rope, wk, bk, kn, out + BSC, 1);
    qkv_rmsnorm_rope_kernel<<<grid, block, 0, stream>>>(x, rope, wv, bv, qn, out + 2 * BSC, 0);
}